// GraphConvolution_6098853560838
// MI455X (gfx1250) — compile-verified
//
#include <hip/hip_runtime.h>

typedef __attribute__((ext_vector_type(2))) float v2f;
typedef __attribute__((ext_vector_type(8))) float v8f;

#define N_NODES 50000
#define N_EDGES 1000000
#define IN_F    128
#define OUT_F   64

// ---------------------------------------------------------------------------
// Kernel 1: support = features @ W via V_WMMA_F32_16X16X4_F32 (fp32 WMMA).
// 4 waves / block; each wave computes one 16-row M tile x all 4 N tiles.
// A fragment (16x4): lane = h*16+l holds A[row=l][k0+2h] in v0, [k0+2h+1] in v1.
// B fragment (4x16): lane = h*16+l holds W[k0+2h][n] in v0, W[k0+2h+1][n] in v1.
// C/D (16x16 f32): VGPR j holds row (j + 8h), col l.
// ---------------------------------------------------------------------------
__global__ __launch_bounds__(128)
void gcn_gemm_wmma(const float* __restrict__ A, const float* __restrict__ W,
                   float* __restrict__ S) {
    const int lane  = threadIdx.x & 31;
    const int wave  = threadIdx.x >> 5;
    const int tileM = blockIdx.x * 4 + wave;
    if (tileM * 16 >= N_NODES) return;          // whole-wave exit: EXEC stays all-1 for WMMA

    const int h = lane >> 4;                    // half-wave: 0 or 1
    const int l = lane & 15;
    const float* __restrict__ arow = A + (long)(tileM * 16 + l) * IN_F;

    v8f acc0 = {}, acc1 = {}, acc2 = {}, acc3 = {};

    #pragma unroll 4
    for (int k0 = 0; k0 < IN_F; k0 += 4) {
        const int ka = k0 + 2 * h;
        v2f a = *(const v2f*)(arow + ka);              // A[row][ka], A[row][ka+1]

        const float* __restrict__ w0 = W + ka * OUT_F + l;       // row ka
        const float* __restrict__ w1 = w0 + OUT_F;               // row ka+1
        v2f b0 = { w0[0],  w1[0]  };                   // N tile 0
        v2f b1 = { w0[16], w1[16] };                   // N tile 1
        v2f b2 = { w0[32], w1[32] };                   // N tile 2
        v2f b3 = { w0[48], w1[48] };                   // N tile 3

        acc0 = __builtin_amdgcn_wmma_f32_16x16x4_f32(false, a, false, b0, (short)0, acc0, false, false);
        acc1 = __builtin_amdgcn_wmma_f32_16x16x4_f32(false, a, false, b1, (short)0, acc1, false, false);
        acc2 = __builtin_amdgcn_wmma_f32_16x16x4_f32(false, a, false, b2, (short)0, acc2, false, false);
        acc3 = __builtin_amdgcn_wmma_f32_16x16x4_f32(false, a, false, b3, (short)0, acc3, false, false);
    }

    // Store: acc*[j] is row M = tileM*16 + j + 8h, col = l (+16*ntile)
    float* __restrict__ srow = S + (long)(tileM * 16 + 8 * h) * OUT_F + l;
    #pragma unroll
    for (int j = 0; j < 8; ++j) {
        float* p = srow + j * OUT_F;
        p[0]  = acc0[j];
        p[16] = acc1[j];
        p[32] = acc2[j];
        p[48] = acc3[j];
    }
}

// ---------------------------------------------------------------------------
// Kernel 2: out[i] = b[i % 64]  (bias broadcast; re-inits poisoned d_out)
// ---------------------------------------------------------------------------
__global__ __launch_bounds__(256)
void gcn_bias_init(float* __restrict__ out, const float* __restrict__ b) {
    int i = blockIdx.x * 256 + threadIdx.x;
    if (i < N_NODES * OUT_F) out[i] = b[i & (OUT_F - 1)];
}

// ---------------------------------------------------------------------------
// Kernel 3: edge scatter. 64 threads (2 waves) per edge; lane handles one
// feature. support + out are L2-resident (12.8 MB each << 192 MB), so the
// gather and the fire-and-forget global_atomic_add_f32 both run at L2 rates.
// ---------------------------------------------------------------------------
__global__ __launch_bounds__(256)
void gcn_scatter(const float* __restrict__ S, const int* __restrict__ esrc,
                 const int* __restrict__ edst, const float* __restrict__ ew,
                 float* __restrict__ out) {
    int gid = blockIdx.x * 256 + threadIdx.x;
    int e = gid >> 6;                 // edge index (wave-uniform)
    int f = gid & 63;                 // feature index
    if (e >= N_EDGES) return;

    int   s = esrc[e];
    int   d = edst[e];
    float w = ew[e];
    float v = S[(long)s * OUT_F + f] * w;
    // hardware f32 atomic (no return) -> global_atomic_add_f32, tracked by STOREcnt
    unsafeAtomicAdd(&out[(long)d * OUT_F + f], v);
}

// ---------------------------------------------------------------------------
extern "C" void kernel_launch(void* const* d_in, const int* in_sizes, int n_in,
                              void* d_out, int out_size, void* d_ws, size_t ws_size,
                              hipStream_t stream) {
    const float* features = (const float*)d_in[0];   // [50000,128]
    const int*   edge_src = (const int*)  d_in[1];   // [1M]
    const int*   edge_dst = (const int*)  d_in[2];   // [1M]
    const float* edge_w   = (const float*)d_in[3];   // [1M]
    const float* W        = (const float*)d_in[4];   // [128,64]
    const float* b        = (const float*)d_in[5];   // [64]
    float* out = (float*)d_out;                      // [50000,64]
    float* S   = (float*)d_ws;                       // support scratch: 12.8 MB

    const int mtiles      = N_NODES / 16;            // 3125 (exact)
    const int gemm_blocks = (mtiles + 3) / 4;        // 782

    gcn_gemm_wmma<<<gemm_blocks, 128, 0, stream>>>(features, W, S);

    const int bias_blocks = (N_NODES * OUT_F + 255) / 256;
    gcn_bias_init<<<bias_blocks, 256, 0, stream>>>(out, b);

    const int scat_blocks = (N_EDGES * 64) / 256;    // 250000
    gcn_scatter<<<scat_blocks, 256, 0, stream>>>(S, edge_src, edge_dst, edge_w, out);
}